// FrustumPointnet_58016418234388
// MI455X (gfx1250) — compile-verified
//
#include <hip/hip_runtime.h>
#include <hip/hip_bf16.h>

// ---------------------------------------------------------------------------
// FrustumPointnet point_cloud_masking for MI455X (gfx1250, wave32).
// B=64, N=131072, C=4, M=2048.
// Memory-bound: pass1 streams pc+logits (192 MB) and emits a 1 MB mask
// bitmask; compaction re-reads only the bitmask. ~266 MB total -> ~11.5us
// at 23.3 TB/s. Cross-lane reductions run on the matrix unit
// (v_wmma_f32_16x16x4_f32).
// ---------------------------------------------------------------------------

#define BB      64
#define NN      131072
#define MM      2048
#define SEGS    128
#define SEGLEN  (NN / SEGS)   // 1024
#define TPB     256
#define NWAVES  (TPB / 32)
#define NWORDS  (NN / 32)     // mask words per batch

typedef __attribute__((ext_vector_type(2))) float v2f;
typedef __attribute__((ext_vector_type(8))) float v8f;

// ---------------------------------------------------------------------------
// Wave32 reduction of four f32 accumulators using the WMMA unit.
// Stage 1 (2 WMMAs): one-hot B selector columns give
//   D[m][q] = part_q[lane m] + part_q[lane m+16], q=0..3.
// Stage 2 (4 WMMAs): D's VGPR pairs fed back as A against all-ones B,
// accumulating the 16 remaining rows: D2[q][n] = full 32-lane total of
// quantity q (lanes 0..15, vector elements 0..3).
// ---------------------------------------------------------------------------
__device__ inline v8f wave_reduce4_wmma(float sw, float sx, float sy, float sz,
                                        int lane) {
  const int c = lane & 15;
  v2f aWX = {sw, sx};
  v2f aYZ = {sy, sz};
  v2f b1  = {c == 0 ? 1.0f : 0.0f, c == 1 ? 1.0f : 0.0f};
  v2f b2  = {c == 2 ? 1.0f : 0.0f, c == 3 ? 1.0f : 0.0f};
  v8f d = {0.f,0.f,0.f,0.f,0.f,0.f,0.f,0.f};
  d = __builtin_amdgcn_wmma_f32_16x16x4_f32(false, aWX, false, b1, (short)0, d, false, false);
  d = __builtin_amdgcn_wmma_f32_16x16x4_f32(false, aYZ, false, b2, (short)0, d, false, false);

  v2f ones = {1.0f, 1.0f};
  v2f a01 = {d[0], d[1]};
  v2f a23 = {d[2], d[3]};
  v2f a45 = {d[4], d[5]};
  v2f a67 = {d[6], d[7]};
  v8f d2 = {0.f,0.f,0.f,0.f,0.f,0.f,0.f,0.f};
  d2 = __builtin_amdgcn_wmma_f32_16x16x4_f32(false, a01, false, ones, (short)0, d2, false, false);
  d2 = __builtin_amdgcn_wmma_f32_16x16x4_f32(false, a23, false, ones, (short)0, d2, false, false);
  d2 = __builtin_amdgcn_wmma_f32_16x16x4_f32(false, a45, false, ones, (short)0, d2, false, false);
  d2 = __builtin_amdgcn_wmma_f32_16x16x4_f32(false, a67, false, ones, (short)0, d2, false, false);
  return d2;  // lanes 0..15: d2[0]=count, d2[1]=sumX, d2[2]=sumY, d2[3]=sumZ
}

// ---------------------------------------------------------------------------
// threefry2x32-20 matching JAX (key = PRNGKey(42) = {0, 42}).
// ---------------------------------------------------------------------------
__device__ inline void threefry2x32(unsigned k0, unsigned k1,
                                    unsigned c0, unsigned c1,
                                    unsigned* o0, unsigned* o1) {
  unsigned ks[3] = {k0, k1, k0 ^ k1 ^ 0x1BD11BDAu};
  unsigned x0 = c0 + ks[0];
  unsigned x1 = c1 + ks[1];
  const unsigned rot[2][4] = {{13u,15u,26u,6u},{17u,29u,16u,24u}};
  #pragma unroll
  for (int i = 0; i < 5; ++i) {
    const unsigned* r = rot[i & 1];
    #pragma unroll
    for (int j = 0; j < 4; ++j) {
      x0 += x1;
      x1 = (x1 << r[j]) | (x1 >> (32u - r[j]));
      x1 ^= x0;
    }
    x0 += ks[(i + 1) % 3];
    x1 += ks[(i + 2) % 3] + (unsigned)(i + 1);
  }
  *o0 = x0; *o1 = x1;
}

// ---------------------------------------------------------------------------
// K0: zero the per-batch accumulators (d_ws is poisoned, not zeroed).
// ---------------------------------------------------------------------------
__global__ __launch_bounds__(TPB) void k_init(float* __restrict__ sums) {
  int i = blockIdx.x * TPB + threadIdx.x;
  if (i < BB * 4) sums[i] = 0.0f;
}

// ---------------------------------------------------------------------------
// K1: streaming pass. Each block = one 1024-point segment of one batch.
// Computes mask, ballots it into a per-wave 32-bit word (one word = 32
// consecutive points, bit = lane = point), accumulates {count, sum xyz};
// block reduction goes wave-WMMA -> LDS -> global f32 atomics.
// ---------------------------------------------------------------------------
__global__ __launch_bounds__(TPB) void k_reduce(const float* __restrict__ pc,
                                                const float* __restrict__ lg,
                                                float* __restrict__ sums,
                                                int* __restrict__ segCnt,
                                                unsigned* __restrict__ maskBits) {
  const int s = blockIdx.x;
  const int b = blockIdx.y;
  const int tid = threadIdx.x;
  const int lane = tid & 31;
  const size_t pcBase = (size_t)b * NN;        // float4 units
  const size_t lgBase = (size_t)b * 2u * NN;
  const float4* __restrict__ pc4 = (const float4*)pc;

  float sw = 0.f, sx = 0.f, sy = 0.f, sz = 0.f;
  const int n0 = s * SEGLEN;
  #pragma unroll
  for (int k = 0; k < SEGLEN / TPB; ++k) {
    int n = n0 + k * TPB + tid;
    if (k + 1 < SEGLEN / TPB) {                     // global_prefetch_b8
      __builtin_prefetch(&pc4[pcBase + n + TPB], 0, 0);
      __builtin_prefetch(&lg[lgBase + n + TPB], 0, 0);
    }
    float l0 = lg[lgBase + n];
    float l1 = lg[lgBase + NN + n];
    float4 p = pc4[pcBase + n];
    bool msk = l0 < l1;
    unsigned long long bal = __ballot((int)msk);
    if (lane == 0)   // n is the wave-base point index for lane 0
      maskBits[(size_t)b * NWORDS + (unsigned)(n >> 5)] = (unsigned)bal;
    float m = msk ? 1.0f : 0.0f;
    sw += m;
    sx += m * p.x;
    sy += m * p.y;
    sz += m * p.z;
  }

  v8f tot = wave_reduce4_wmma(sw, sx, sy, sz, lane);

  __shared__ float wsum[NWAVES][4];
  if (lane == 0) {
    int wv = tid >> 5;
    wsum[wv][0] = tot[0]; wsum[wv][1] = tot[1];
    wsum[wv][2] = tot[2]; wsum[wv][3] = tot[3];
  }
  __syncthreads();
  if (tid == 0) {
    float a0 = 0.f, a1 = 0.f, a2 = 0.f, a3 = 0.f;
    #pragma unroll
    for (int w = 0; w < NWAVES; ++w) {
      a0 += wsum[w][0]; a1 += wsum[w][1]; a2 += wsum[w][2]; a3 += wsum[w][3];
    }
    atomicAdd(&sums[b * 4 + 0], a0);
    atomicAdd(&sums[b * 4 + 1], a1);
    atomicAdd(&sums[b * 4 + 2], a2);
    atomicAdd(&sums[b * 4 + 3], a3);
    segCnt[b * SEGS + s] = (int)(a0 + 0.5f);
  }
}

// ---------------------------------------------------------------------------
// K2: per-batch exclusive scan over segment counts; writes total count and
// the masked xyz mean (into the tail of d_out).
// ---------------------------------------------------------------------------
__global__ __launch_bounds__(SEGS) void k_scan(const float* __restrict__ sums,
                                               const int* __restrict__ segCnt,
                                               int* __restrict__ segOff,
                                               int* __restrict__ cntBuf,
                                               float* __restrict__ meanOut) {
  const int b = blockIdx.x;
  const int tid = threadIdx.x;
  __shared__ int sc[SEGS];
  int v = segCnt[b * SEGS + tid];
  sc[tid] = v;
  __syncthreads();
  for (int off = 1; off < SEGS; off <<= 1) {
    int t = (tid >= off) ? sc[tid - off] : 0;
    __syncthreads();
    sc[tid] += t;
    __syncthreads();
  }
  segOff[b * SEGS + tid] = sc[tid] - v;
  if (tid == SEGS - 1) {
    int total = sc[tid];
    cntBuf[b] = total;
    float c = fmaxf((float)total, 1.0f);
    meanOut[b * 3 + 0] = sums[b * 4 + 1] / c;
    meanOut[b * 3 + 1] = sums[b * 4 + 2] / c;
    meanOut[b * 3 + 2] = sums[b * 4 + 3] / c;
  }
}

// ---------------------------------------------------------------------------
// K3: stable compaction of masked indices driven entirely by the 1 MB mask
// bitmask (no logits re-read). Each wave's 32 points map to one word:
// bit = lane, rank = popcount of lower bits -> stable order for free.
// ---------------------------------------------------------------------------
__global__ __launch_bounds__(TPB) void k_compact(const unsigned* __restrict__ maskBits,
                                                 const int* __restrict__ segOff,
                                                 int* __restrict__ compactIdx) {
  const int s = blockIdx.x;
  const int b = blockIdx.y;
  const int tid = threadIdx.x;
  const size_t cBase = (size_t)b * NN;
  const int lane = tid & 31;
  const int wv = tid >> 5;

  __shared__ int wcnt[NWAVES];
  int base = segOff[b * SEGS + s];

  #pragma unroll
  for (int c = 0; c < SEGLEN / TPB; ++c) {
    int n = s * SEGLEN + c * TPB + tid;
    unsigned word = maskBits[(size_t)b * NWORDS + (unsigned)(n >> 5)];
    bool m = (word >> lane) & 1u;
    int rank = __popc(word & ((1u << lane) - 1u));
    if (lane == 0) wcnt[wv] = __popc(word);
    __syncthreads();
    int wbase = 0, tot = 0;
    #pragma unroll
    for (int i = 0; i < NWAVES; ++i) {
      int vi = wcnt[i];
      tot += vi;
      if (i < wv) wbase += vi;
    }
    if (m) compactIdx[cBase + base + wbase + rank] = n;
    base += tot;
    __syncthreads();
  }
}

// ---------------------------------------------------------------------------
// K4: selection + centering. sel = j for j<cnt (when cnt<=M), else a
// threefry-drawn rank in [0, cnt); empty batches emit the batch index.
// ---------------------------------------------------------------------------
__global__ __launch_bounds__(TPB) void k_select(const float* __restrict__ pc,
                                                const int* __restrict__ compactIdx,
                                                const int* __restrict__ cntBuf,
                                                const float* __restrict__ meanOut,
                                                float* __restrict__ obj) {
  const int b = blockIdx.y;
  const int j = blockIdx.x * TPB + threadIdx.x;   // 0..MM-1
  const int c = cntBuf[b];
  float ox, oy, oz;
  if (c == 0) {
    ox = oy = oz = (float)b;
  } else {
    unsigned gid = (unsigned)(b * MM + j);
    const unsigned half = (unsigned)(BB * MM) / 2u;   // JAX split-counter layout
    unsigned c0 = (gid < half) ? gid : gid - half;
    unsigned c1 = c0 + half;
    unsigned x0, x1;
    threefry2x32(0u, 42u, c0, c1, &x0, &x1);
    unsigned bits = (gid < half) ? x0 : x1;
    int r = (int)(bits % (unsigned)c);
    int sel = (j < c && c <= MM) ? j : r;
    int idx = compactIdx[(size_t)b * NN + sel];
    const float4* __restrict__ pc4 = (const float4*)pc;
    float4 p = pc4[(size_t)b * NN + idx];
    ox = p.x - meanOut[b * 3 + 0];
    oy = p.y - meanOut[b * 3 + 1];
    oz = p.z - meanOut[b * 3 + 2];
  }
  size_t o = ((size_t)b * MM + j) * 3u;
  obj[o + 0] = ox;
  obj[o + 1] = oy;
  obj[o + 2] = oz;
}

// ---------------------------------------------------------------------------
// Host-side launcher (graph-capture safe: no alloc/sync, stream-only).
// ---------------------------------------------------------------------------
extern "C" void kernel_launch(void* const* d_in, const int* in_sizes, int n_in,
                              void* d_out, int out_size, void* d_ws, size_t ws_size,
                              hipStream_t stream) {
  (void)in_sizes; (void)n_in; (void)out_size; (void)ws_size;
  const float* pc = (const float*)d_in[0];   // (B, N, 4)
  const float* lg = (const float*)d_in[1];   // (B, 2, N)
  float* obj = (float*)d_out;                     // (B, M, 3)
  float* meanOut = obj + (size_t)BB * MM * 3;     // (B, 3)

  // Workspace layout (~34.6 MB):
  float* sums = (float*)d_ws;                     // B*4 f32 accumulators
  int* cntBuf = (int*)(sums + BB * 4);            // B
  int* segCnt = cntBuf + BB;                      // B*SEGS
  int* segOff = segCnt + BB * SEGS;               // B*SEGS
  unsigned* maskBits = (unsigned*)(segOff + BB * SEGS);  // B*N/32 mask words
  int* compactIdx = (int*)(maskBits + (size_t)BB * NWORDS); // B*N index lists

  k_init<<<1, TPB, 0, stream>>>(sums);
  k_reduce<<<dim3(SEGS, BB), TPB, 0, stream>>>(pc, lg, sums, segCnt, maskBits);
  k_scan<<<dim3(BB), SEGS, 0, stream>>>(sums, segCnt, segOff, cntBuf, meanOut);
  k_compact<<<dim3(SEGS, BB), TPB, 0, stream>>>(maskBits, segOff, compactIdx);
  k_select<<<dim3(MM / TPB, BB), TPB, 0, stream>>>(pc, compactIdx, cntBuf, meanOut, obj);
}